// LengthRegulatorTRT_45904610460089
// MI455X (gfx1250) — compile-verified
//
#include <hip/hip_runtime.h>

#define B_ 32
#define S_ 256
#define H_ 384
#define M_ 2000
#define ROWS_ (B_ * M_)
#define WPB_ 8               // waves per block (wave32)
#define TPB_ (WPB_ * 32)

#if defined(__gfx1250__) &&                                         \
    __has_builtin(__builtin_amdgcn_global_load_async_to_lds_b128) && \
    __has_builtin(__builtin_amdgcn_global_store_async_from_lds_b128) && \
    __has_builtin(__builtin_amdgcn_s_wait_asynccnt)
#define HAVE_ASYNC_LDS 1
#else
#define HAVE_ASYNC_LDS 0
#endif

// ---------------------------------------------------------------------------
// Kernel 2 (defined first so it leads the disassembly): one wave32 per output
// row [b, m, 0:384]. idxmap holds the *global* source row (b*S + s), or -1.
// Branchless: srow < 0 selects a zeroed 1536 B row in workspace, then every
// wave runs the same async mem->LDS->mem copy (3 x 32 lanes x B128).
// ---------------------------------------------------------------------------
__global__ __launch_bounds__(TPB_) void lr_gather_rows(
    const float* __restrict__ x,
    const int* __restrict__ idxmap,
    const float* __restrict__ zrow,     // 384 zero floats, 16B-aligned
    float* __restrict__ out)
{
#if HAVE_ASYNC_LDS
    __shared__ float lds_buf[WPB_ * H_];   // 12 KB: one row staging per wave
#endif
    const int tid  = threadIdx.x;
    const int wave = tid >> 5;
    const int lane = tid & 31;
    const int row  = blockIdx.x * WPB_ + wave;   // ROWS_ % WPB_ == 0, no tail

    // srow is wave-uniform; readfirstlane -> scalar select of the source base.
    const int srow = __builtin_amdgcn_readfirstlane(idxmap[row]);

    const float* sbase = (srow >= 0) ? (x + (size_t)srow * H_) : zrow;
    const float* src   = sbase + lane * 4;
    float*       dst   = out + (size_t)row * H_ + lane * 4;

#if HAVE_ASYNC_LDS
    typedef int v4i __attribute__((ext_vector_type(4)));
    typedef __attribute__((address_space(1))) v4i* gptr;   // global 16B chunk
    typedef __attribute__((address_space(3))) v4i* lptr;   // LDS 16B chunk
    float* lbase = &lds_buf[wave * H_ + lane * 4];

    #pragma unroll
    for (int i = 0; i < 3; ++i)
        __builtin_amdgcn_global_load_async_to_lds_b128(
            (gptr)(src + i * 128), (lptr)(lbase + i * 128), 0, 0);

    __builtin_amdgcn_s_wait_asynccnt(0);         // loads landed in LDS

    #pragma unroll
    for (int i = 0; i < 3; ++i)
        __builtin_amdgcn_global_store_async_from_lds_b128(
            (gptr)(dst + i * 128), (lptr)(lbase + i * 128), 0, 0);
    // ASYNCcnt drains via S_ENDPGM's implicit wait-idle.
#else
    #pragma unroll
    for (int i = 0; i < 3; ++i)
        *(float4*)(dst + i * 128) = *(const float4*)(src + i * 128);
#endif
}

// ---------------------------------------------------------------------------
// Kernel 1: per-batch inclusive scan of durations; build frame -> source-row
// map (b*S + s, or -1 past the end); zero the shared zero-row; emit mel_len
// (int32, appended after the float output).
// ---------------------------------------------------------------------------
__global__ __launch_bounds__(S_) void lr_build_index_map(
    const float* __restrict__ duration,
    int* __restrict__ idxmap,
    float* __restrict__ zrow,
    int* __restrict__ mel_len_out)
{
    __shared__ int scan[S_];
    const int b = blockIdx.x;
    const int s = threadIdx.x;

    if (b == 0) {                      // block 0 also zeroes the zero-row
        zrow[s] = 0.f;
        if (s < H_ - S_) zrow[S_ + s] = 0.f;
    }

    const int d = (int)duration[b * S_ + s];   // durations are exact small ints
    scan[s] = d;
    __syncthreads();

    // Hillis-Steele inclusive scan over 256 elements (8 steps)
    #pragma unroll
    for (int off = 1; off < S_; off <<= 1) {
        const int v   = scan[s];
        const int add = (s >= off) ? scan[s - off] : 0;
        __syncthreads();
        scan[s] = v + add;
        __syncthreads();
    }

    const int end   = scan[s];
    const int start = end - d;
    const int total = scan[S_ - 1];

    // frames belonging to phoneme s (clamped to M_): store global source row
    const int e    = (end < M_) ? end : M_;
    const int srow = b * S_ + s;
    for (int m = start; m < e; ++m)
        idxmap[b * M_ + m] = srow;

    // frames past the last phoneme -> -1 (zero output)
    for (int m = total + s; m < M_; m += S_)
        idxmap[b * M_ + m] = -1;

    if (s == 0)
        mel_len_out[b] = (total < M_) ? total : M_;
}

// ---------------------------------------------------------------------------
extern "C" void kernel_launch(void* const* d_in, const int* in_sizes, int n_in,
                              void* d_out, int out_size, void* d_ws, size_t ws_size,
                              hipStream_t stream) {
    (void)in_sizes; (void)n_in; (void)out_size; (void)ws_size;

    const float* x        = (const float*)d_in[0];   // [32, 256, 384] f32
    const float* duration = (const float*)d_in[1];   // [32, 256] f32

    float* out     = (float*)d_out;                          // [32, 2000, 384] f32
    int*   mel_len = (int*)(out + (size_t)B_ * M_ * H_);     // [32] i32 (raw bits)
    int*   idxmap  = (int*)d_ws;                             // [32, 2000] i32
    float* zrow    = (float*)d_ws + ROWS_;                   // 384 f32, 16B-aligned

    lr_build_index_map<<<B_, S_, 0, stream>>>(duration, idxmap, zrow, mel_len);
    lr_gather_rows<<<ROWS_ / WPB_, TPB_, 0, stream>>>(x, idxmap, zrow, out);
}